// ExpertCapacityBuffer_80444737454353
// MI455X (gfx1250) — compile-verified
//
#include <hip/hip_runtime.h>
#include <hip/hip_bf16.h>
#include <math.h>

#define WAVE_SZ    32
#define NEXPERTS   64
#define CNT_BLOCK  256
#define SCAN_BLOCK 256
#define MAX_SEGS   8192   // 2N/32 for N=131072; fits 32KB LDS of the 320KB WGP pool

// ---- wave32 match_any emulation: mask of active lanes whose 6-bit expert
// ---- equals mine, built from 6 ballots (CDNA5 is wave32-only).
__device__ __forceinline__ unsigned match_expert(int e, unsigned act) {
  unsigned m = act;
#pragma unroll
  for (int b = 0; b < 6; ++b) {
    unsigned bal = (unsigned)__ballot((e >> b) & 1);
    m &= ((e >> b) & 1) ? bal : ~bal;
  }
  return m;
}

// TOP_K == 2, slot-major flat order: i = slot*N + token.
// slot = (i >= N) — avoids the runtime integer division the compiler
// otherwise expands into a ~20-op reciprocal sequence per lane.
__device__ __forceinline__ int pair_addr(int i, int N) {
  int slot  = (i >= N) ? 1 : 0;
  int token = i - (slot ? N : 0);
  return token * 2 + slot;
}

// ---------------------------------------------------------------------------
// K1: per-32-element-segment, per-expert histogram.  counts[seg][e].
// ---------------------------------------------------------------------------
__global__ void k_seg_count(const int* __restrict__ eidx, int* __restrict__ counts,
                            int N, int totalFlat, int nseg) {
  int i    = blockIdx.x * CNT_BLOCK + threadIdx.x;
  int lane = threadIdx.x & (WAVE_SZ - 1);
  int seg  = i >> 5;
  bool valid = i < totalFlat;
  int e = 0;
  if (valid) e = eidx[pair_addr(i, N)] & (NEXPERTS - 1);

  unsigned act = (unsigned)__ballot(valid);
  if (act != 0u && seg < nseg) {           // zero-fill all 64 expert slots (2/lane)
    counts[seg * NEXPERTS + lane]           = 0;
    counts[seg * NEXPERTS + lane + WAVE_SZ] = 0;
  }
  unsigned m = match_expert(e, act);
  if (valid) {
    int leader = 31 - __clz((int)m);       // highest lane sharing my expert
    if (lane == leader)
      counts[seg * NEXPERTS + e] = __popc(m);   // same-wave stores stay in order
  }
}

// ---------------------------------------------------------------------------
// K2: per-expert exclusive scan across segments (in place).
// One block per expert.  Strided column gather staged into LDS via the
// gfx1250 async global->LDS path (ASYNCcnt).
// ---------------------------------------------------------------------------
__global__ void k_col_scan(int* __restrict__ counts, int nseg) {
  __shared__ int col[MAX_SEGS];
  __shared__ int psum[SCAN_BLOCK];
  const int e = blockIdx.x;
  const int t = threadIdx.x;

  for (int s = t; s < nseg; s += SCAN_BLOCK) {
#if defined(__HIP_DEVICE_COMPILE__) && __has_builtin(__builtin_amdgcn_global_load_async_to_lds_b32)
    __builtin_amdgcn_global_load_async_to_lds_b32(
        (__attribute__((address_space(1))) int*)(counts + s * NEXPERTS + e),
        (__attribute__((address_space(3))) int*)(col + s),
        /*offset=*/0, /*cpol=*/0);
#else
    col[s] = counts[s * NEXPERTS + e];
#endif
  }
#if defined(__HIP_DEVICE_COMPILE__) && __has_builtin(__builtin_amdgcn_global_load_async_to_lds_b32)
#if __has_builtin(__builtin_amdgcn_s_wait_asynccnt)
  __builtin_amdgcn_s_wait_asynccnt(0);
#else
  asm volatile("s_wait_asynccnt 0" ::: "memory");
#endif
#endif
  __syncthreads();

  int spt = (nseg + SCAN_BLOCK - 1) / SCAN_BLOCK;
  int s0 = t * spt;       if (s0 > nseg) s0 = nseg;
  int s1 = s0 + spt;      if (s1 > nseg) s1 = nseg;

  int sum = 0;
  for (int s = s0; s < s1; ++s) sum += col[s];
  psum[t] = sum;
  __syncthreads();

  int v = sum;                                   // Hillis–Steele inclusive scan
  for (int d = 1; d < SCAN_BLOCK; d <<= 1) {
    int x = (t >= d) ? psum[t - d] : 0;
    __syncthreads();
    v += x;
    psum[t] = v;
    __syncthreads();
  }
  int run = v - sum;                             // exclusive base for my chunk
  for (int s = s0; s < s1; ++s) {
    int c = col[s];
    counts[s * NEXPERTS + e] = run;              // overwrite with exclusive prefix
    run += c;
  }
}

// ---------------------------------------------------------------------------
// K3: final rank, capacity test, emit capped weights + indices(as float).
// ---------------------------------------------------------------------------
__global__ void ExpertCapacityBuffer_80444737454353_kernel(
    const float* __restrict__ wts, const int* __restrict__ eidx,
    const int* __restrict__ offs, float* __restrict__ outW,
    float* __restrict__ outI, int N, int totalFlat, int capacity) {
  int i    = blockIdx.x * CNT_BLOCK + threadIdx.x;
  int lane = threadIdx.x & (WAVE_SZ - 1);
  bool valid = i < totalFlat;
  int e = 0, a = 0;
  float w = 0.0f;
  if (valid) {
    a = pair_addr(i, N);
    __builtin_prefetch(wts + a + 4096, 0, 0);    // global_prefetch_b8
    e = eidx[a] & (NEXPERTS - 1);
    w = wts[a];
  }
  unsigned act = (unsigned)__ballot(valid);
  unsigned m   = match_expert(e, act);
  unsigned lt  = (lane == 0) ? 0u : (0xFFFFFFFFu >> (32 - lane));
  int prior    = __popc(m & lt);
  if (valid) {
    int seg  = i >> 5;
    int rank = offs[seg * NEXPERTS + e] + prior;
    outW[a] = (rank < capacity) ? w : 0.0f;
    outI[a] = (float)e;
  }
}

// ---------------------------------------------------------------------------
// K4: overflow mask: both slots capped to zero.
// ---------------------------------------------------------------------------
__global__ void k_overflow(const float* __restrict__ outW, float* __restrict__ outM, int N) {
  int t = blockIdx.x * 256 + threadIdx.x;
  if (t < N) {
    float s = outW[2 * t] + outW[2 * t + 1];
    outM[t] = (s == 0.0f) ? 1.0f : 0.0f;
  }
}

extern "C" void kernel_launch(void* const* d_in, const int* in_sizes, int n_in,
                              void* d_out, int out_size, void* d_ws, size_t ws_size,
                              hipStream_t stream) {
  const float* wts  = (const float*)d_in[0];   // (N, 2) float32
  const int*   eidx = (const int*)d_in[1];     // (N, 2) integer expert ids
  (void)n_in; (void)out_size; (void)ws_size;   // n_tokens scalar (d_in[2]) derivable

  const int totalFlat = in_sizes[0];           // N * TOP_K
  const int N         = totalFlat / 2;
  const int nseg      = (totalFlat + WAVE_SZ - 1) / WAVE_SZ;   // 8192 for N=131072
  int capacity = (int)ceil(1.25 * (double)totalFlat / (double)NEXPERTS);
  if (capacity < 1) capacity = 1;

  int*   counts = (int*)d_ws;                  // nseg * 64 * 4 B = 2 MB
  float* outW   = (float*)d_out;               // [0, 2N): capped weights
  float* outI   = outW + (size_t)2 * N;        // [2N, 4N): indices as float
  float* outM   = outW + (size_t)4 * N;        // [4N, 5N): overflow mask

  const int blocks = (totalFlat + CNT_BLOCK - 1) / CNT_BLOCK;

  k_seg_count<<<blocks, CNT_BLOCK, 0, stream>>>(eidx, counts, N, totalFlat, nseg);
  k_col_scan<<<NEXPERTS, SCAN_BLOCK, 0, stream>>>(counts, nseg);
  ExpertCapacityBuffer_80444737454353_kernel<<<blocks, CNT_BLOCK, 0, stream>>>(
      wts, eidx, counts, outW, outI, N, totalFlat, capacity);
  k_overflow<<<(N + 255) / 256, 256, 0, stream>>>(outW, outM, N);
}